// GlobalNodeInductiveActorNetwork_29729763623416
// MI455X (gfx1250) — compile-verified
//
#include <hip/hip_runtime.h>
#include <math.h>

typedef __attribute__((ext_vector_type(2))) float v2f;
typedef __attribute__((ext_vector_type(8))) float v8f;

#define N_TOTAL 65536
#define NODES_PER_GRAPH 1024
#define HOSTS 512
#define BGRAPH 64
#define IN_DIM 256
#define H1 256
#define H2 64
#define GDIM 256
#define N_ACT 11
#define N_EDGES 524288
#define M_HOSTS 32768   /* BGRAPH*HOSTS */

// ---------------------------------------------------------------------------
// Generic fp32 WMMA GEMM:  Y[m,n] = epi( sum_k X[g(m),k]*W[k,n] )
// block = 128 threads (4 waves); wave -> 16x64 tile; grid = (N/64, M/64)
// gather: 0 identity rows, 1 host-row mapping m -> (m>>9)*1024 + (m&511)
// epilogue: +bias (if non-null), *rowscale[m] (if non-null), relu (if flag)
// Uses V_WMMA_F32_16X16X4_F32 (native fp32 matrix op on gfx1250).
// ---------------------------------------------------------------------------
__global__ void k_gemm_wmma(const float* __restrict__ X,
                            const float* __restrict__ Wm,
                            const float* __restrict__ bias,
                            const float* __restrict__ rowscale,
                            float* __restrict__ Y,
                            int N, int K, int gather, int relu) {
  const int lane  = threadIdx.x & 31;
  const int wave  = threadIdx.x >> 5;
  const int row0  = blockIdx.y * 64 + wave * 16;
  const int col0  = blockIdx.x * 64;
  const int mrow  = row0 + (lane & 15);
  const int koff  = (lane >> 4) << 1;   // 0 or 2 (A/B K sub-offset per ISA layout)
  const int nlane = lane & 15;

  const int srow = gather ? (((mrow >> 9) << 10) | (mrow & 511)) : mrow;
  const float* __restrict__ xrow = X + (size_t)srow * K;

  v8f acc0 = {0.f,0.f,0.f,0.f,0.f,0.f,0.f,0.f};
  v8f acc1 = acc0, acc2 = acc0, acc3 = acc0;

  for (int k = 0; k < K; k += 4) {
    // A fragment: A[M=lane&15][k+koff .. k+koff+1], contiguous -> one b64 load
    v2f a = *(const v2f*)(xrow + k + koff);
    const float* __restrict__ wp = Wm + (size_t)(k + koff) * N + col0 + nlane;
    v2f b0, b1, b2, b3;
    b0.x = wp[0];  b0.y = wp[N];
    b1.x = wp[16]; b1.y = wp[N + 16];
    b2.x = wp[32]; b2.y = wp[N + 32];
    b3.x = wp[48]; b3.y = wp[N + 48];
    acc0 = __builtin_amdgcn_wmma_f32_16x16x4_f32(false, a, false, b0, (short)0, acc0, false, false);
    acc1 = __builtin_amdgcn_wmma_f32_16x16x4_f32(false, a, false, b1, (short)0, acc1, false, false);
    acc2 = __builtin_amdgcn_wmma_f32_16x16x4_f32(false, a, false, b2, (short)0, acc2, false, false);
    acc3 = __builtin_amdgcn_wmma_f32_16x16x4_f32(false, a, false, b3, (short)0, acc3, false, false);
  }

  const int rbase = (lane >> 4) << 3;   // C/D: lanes 16-31 hold M+8 rows
  v8f accs[4] = {acc0, acc1, acc2, acc3};
  for (int j = 0; j < 4; ++j) {
    const int col = col0 + j * 16 + nlane;
    const float bv = bias ? bias[col] : 0.0f;
    for (int r = 0; r < 8; ++r) {
      const int row = row0 + rbase + r;
      float v = accs[j][r] + bv;
      if (rowscale) v *= rowscale[row];
      if (relu) v = fmaxf(v, 0.0f);
      Y[(size_t)row * N + col] = v;
    }
  }
}

// ---------------------------------------------------------------------------
__global__ void k_fill(float* __restrict__ p, float v, size_t n) {
  size_t i = (size_t)blockIdx.x * blockDim.x + threadIdx.x;
  if (i < n) p[i] = v;
}

__global__ void k_count_deg(const int* __restrict__ dst, float* __restrict__ deg, int nE) {
  int e = blockIdx.x * blockDim.x + threadIdx.x;
  if (e < nE) atomicAdd(&deg[dst[e]], 1.0f);
}

__global__ void k_rsqrt(float* __restrict__ d, int n) {
  int i = blockIdx.x * blockDim.x + threadIdx.x;
  if (i < n) d[i] = rsqrtf(d[i]);   // deg >= 1 always (self loop)
}

// wave per edge: acc[dst] += hs[src]  (H = 64 or 256, multiple of 32)
__global__ void k_edge_agg(const float* __restrict__ hs,
                           const int* __restrict__ src,
                           const int* __restrict__ dst,
                           float* __restrict__ acc, int nE, int H) {
  int e = (int)(((size_t)blockIdx.x * blockDim.x + threadIdx.x) >> 5);
  int lane = threadIdx.x & 31;
  if (e >= nE) return;
  const float* __restrict__ h = hs + (size_t)src[e] * H;
  float* __restrict__ a = acc + (size_t)dst[e] * H;
  for (int k = lane; k < H; k += 32) atomicAdd(&a[k], h[k]);
}

// x[i,k] = relu((acc[i,k] + hs[i,k]) * dinv[i] + b[k]);  H = 1<<Hbits
__global__ void k_gcn_final(const float* __restrict__ acc, const float* __restrict__ hs,
                            const float* __restrict__ dinv, const float* __restrict__ bias,
                            float* __restrict__ xo, size_t total, int Hbits) {
  size_t idx = (size_t)blockIdx.x * blockDim.x + threadIdx.x;
  if (idx >= total) return;
  int i = (int)(idx >> Hbits);
  int k = (int)(idx & ((1u << Hbits) - 1));
  float v = (acc[idx] + hs[idx]) * dinv[i] + bias[k];
  xo[idx] = fmaxf(v, 0.0f);
}

// wave per host row: softmax(A[m,:]) over feature dim, out[g] += softmax*F[m,:]
__global__ void k_attn_reduce(const float* __restrict__ A, const float* __restrict__ F,
                              float* __restrict__ aout, int M, int H) {
  int row  = (int)(((size_t)blockIdx.x * blockDim.x + threadIdx.x) >> 5);
  int lane = threadIdx.x & 31;
  if (row >= M) return;
  const float* __restrict__ a = A + (size_t)row * H;
  const float* __restrict__ f = F + (size_t)row * H;
  const int per = H >> 5;       // 8 (H=256) or 2 (H=64)
  float vals[8];
  float mx = -3.402823466e38f;
  for (int j = 0; j < per; ++j) { vals[j] = a[lane + (j << 5)]; mx = fmaxf(mx, vals[j]); }
  for (int off = 16; off; off >>= 1) mx = fmaxf(mx, __shfl_xor(mx, off, 32));
  float s = 0.0f;
  for (int j = 0; j < per; ++j) { vals[j] = __expf(vals[j] - mx); s += vals[j]; }
  for (int off = 16; off; off >>= 1) s += __shfl_xor(s, off, 32);
  const float inv = 1.0f / s;
  float* __restrict__ o = aout + (size_t)(row >> 9) * H;
  for (int j = 0; j < per; ++j)
    atomicAdd(&o[lane + (j << 5)], vals[j] * inv * f[lane + (j << 5)]);
}

// gnew = gprev + (concat(aout, gprev) @ gW + gb);  gW: [H+GDIM, GDIM]
__global__ void k_g_update(const float* __restrict__ aout, const float* __restrict__ gprev,
                           const float* __restrict__ gW, const float* __restrict__ gb,
                           float* __restrict__ gnew, int H) {
  int idx = blockIdx.x * blockDim.x + threadIdx.x;
  if (idx >= BGRAPH * GDIM) return;
  int b = idx >> 8, n = idx & 255;
  float s = gb[n];
  const float* __restrict__ ao = aout + (size_t)b * H;
  for (int k = 0; k < H; ++k) s += ao[k] * gW[(size_t)k * GDIM + n];
  const float* __restrict__ gp = gprev ? gprev + (size_t)b * GDIM : nullptr;
  if (gp) {
    for (int k = 0; k < GDIM; ++k) s += gp[k] * gW[(size_t)(H + k) * GDIM + n];
    s += gp[n];
  }
  gnew[idx] = s;
}

// hostz[m, 0:64] = x2[hostnode(m), :], hostz[m, 64:320] = g2[m>>9, :]
__global__ void k_build_hostz(const float* __restrict__ x2, const float* __restrict__ g2,
                              float* __restrict__ hz) {
  size_t idx = (size_t)blockIdx.x * blockDim.x + threadIdx.x;
  if (idx >= (size_t)M_HOSTS * (H2 + GDIM)) return;
  int m = (int)(idx / (H2 + GDIM));
  int k = (int)(idx % (H2 + GDIM));
  int node = ((m >> 9) << 10) | (m & 511);
  hz[idx] = (k < H2) ? x2[(size_t)node * H2 + k] : g2[(size_t)(m >> 9) * GDIM + (k - H2)];
}

// act[m,a] = hidden[m,:] @ Wo2[:,a] + bo2[a]
__global__ void k_actions(const float* __restrict__ hidden, const float* __restrict__ Wo2,
                          const float* __restrict__ bo2, float* __restrict__ act) {
  int idx = blockIdx.x * blockDim.x + threadIdx.x;
  if (idx >= M_HOSTS * N_ACT) return;
  int m = idx / N_ACT, a = idx % N_ACT;
  const float* __restrict__ h = hidden + (size_t)m * H1;
  float s = bo2[a];
  for (int k = 0; k < H1; ++k) s += h[k] * Wo2[k * N_ACT + a];
  act[idx] = s;
}

// per-graph softmax over 5632 logits; out[b, 2 + a*512 + h] = softmax(act[b,h,a])
__global__ void k_out_softmax(const float* __restrict__ act, float* __restrict__ out) {
  const int b = blockIdx.x, t = threadIdx.x;
  const int NL = HOSTS * N_ACT;   // 5632
  __shared__ float red[256];
  float mx = -3.402823466e38f;
  for (int j = t; j < NL; j += 256) {
    int a = j >> 9, h = j & 511;
    mx = fmaxf(mx, act[(size_t)(b * HOSTS + h) * N_ACT + a]);
  }
  red[t] = mx; __syncthreads();
  for (int off = 128; off; off >>= 1) { if (t < off) red[t] = fmaxf(red[t], red[t + off]); __syncthreads(); }
  mx = red[0]; __syncthreads();
  float s = 0.0f;
  for (int j = t; j < NL; j += 256) {
    int a = j >> 9, h = j & 511;
    s += __expf(act[(size_t)(b * HOSTS + h) * N_ACT + a] - mx);
  }
  red[t] = s; __syncthreads();
  for (int off = 128; off; off >>= 1) { if (t < off) red[t] += red[t + off]; __syncthreads(); }
  const float inv = 1.0f / red[0];
  float* __restrict__ ob = out + (size_t)b * (N_ACT * HOSTS + 2);
  if (t < 2) ob[t] = 0.0f;
  for (int j = t; j < NL; j += 256) {
    int a = j >> 9, h = j & 511;
    ob[2 + j] = __expf(act[(size_t)(b * HOSTS + h) * N_ACT + a] - mx) * inv;
  }
}

// ---------------------------------------------------------------------------
extern "C" void kernel_launch(void* const* d_in, const int* in_sizes, int n_in,
                              void* d_out, int out_size, void* d_ws, size_t ws_size,
                              hipStream_t stream) {
  const float* x   = (const float*)d_in[0];
  const int*   ei  = (const int*)d_in[1];     // int32: [2, N_EDGES]
  const float* W1  = (const float*)d_in[3];  const float* b1  = (const float*)d_in[4];
  const float* W2  = (const float*)d_in[5];  const float* b2  = (const float*)d_in[6];
  const float* a0W = (const float*)d_in[7];  const float* a0b = (const float*)d_in[8];
  const float* f0W = (const float*)d_in[9];  const float* f0b = (const float*)d_in[10];
  const float* g0W = (const float*)d_in[11]; const float* g0b = (const float*)d_in[12];
  const float* a1W = (const float*)d_in[13]; const float* a1b = (const float*)d_in[14];
  const float* f1W = (const float*)d_in[15]; const float* f1b = (const float*)d_in[16];
  const float* g1W = (const float*)d_in[17]; const float* g1b = (const float*)d_in[18];
  const float* a2W = (const float*)d_in[19]; const float* a2b = (const float*)d_in[20];
  const float* f2W = (const float*)d_in[21]; const float* f2b = (const float*)d_in[22];
  const float* g2W = (const float*)d_in[23]; const float* g2b = (const float*)d_in[24];
  const float* Wo1 = (const float*)d_in[25]; const float* bo1 = (const float*)d_in[26];
  const float* Wo2 = (const float*)d_in[27]; const float* bo2 = (const float*)d_in[28];
  float* out = (float*)d_out;

  const int* src = ei;
  const int* dst = ei + N_EDGES;

  // workspace layout (~194 MB, with region reuse)
  char* ws = (char*)d_ws;
  float* dinv = (float*)ws;                                   // 65536 f
  float* aout = (float*)(ws + (1u << 20));                    // 16384 f
  float* g0   = aout + 16384;
  float* g1   = g0 + 16384;
  float* g2   = g1 + 16384;
  float* big1 = (float*)(ws + (2u << 20));                    // 64 MB: hs1/x1 -> hostz
  float* big2 = (float*)(ws + (2u << 20) + (64u << 20));      // 64 MB: acc1 -> hs2/x2 + acc2
  float* regA = (float*)(ws + (2u << 20) + (128u << 20));     // 32 MB: A0/A1/A2 -> hidden
  float* regF = (float*)(ws + (2u << 20) + (160u << 20));     // 32 MB: F0/F1/F2 -> act
  float* hs2  = big2;
  float* acc2 = big2 + (size_t)N_TOTAL * H2;                  // second 16 MB of big2

  const dim3 blk(256), blkG(128);

  // --- degrees -> dinv -----------------------------------------------------
  k_fill<<<dim3(N_TOTAL / 256), blk, 0, stream>>>(dinv, 1.0f, (size_t)N_TOTAL);
  k_count_deg<<<dim3(N_EDGES / 256), blk, 0, stream>>>(dst, dinv, N_EDGES);
  k_rsqrt<<<dim3(N_TOTAL / 256), blk, 0, stream>>>(dinv, N_TOTAL);

  // --- attn block 0 (input features, g=0) ----------------------------------
  k_gemm_wmma<<<dim3(H1 / 64, M_HOSTS / 64), blkG, 0, stream>>>(x, a0W, a0b, nullptr, regA, H1, IN_DIM, 1, 0);
  k_gemm_wmma<<<dim3(H1 / 64, M_HOSTS / 64), blkG, 0, stream>>>(x, f0W, f0b, nullptr, regF, H1, IN_DIM, 1, 0);
  k_fill<<<dim3(64), blk, 0, stream>>>(aout, 0.0f, (size_t)BGRAPH * H1);
  k_attn_reduce<<<dim3(M_HOSTS * 32 / 256), blk, 0, stream>>>(regA, regF, aout, M_HOSTS, H1);
  k_g_update<<<dim3(BGRAPH * GDIM / 256), blk, 0, stream>>>(aout, nullptr, g0W, g0b, g0, H1);

  // --- GCN layer 1 ---------------------------------------------------------
  k_gemm_wmma<<<dim3(H1 / 64, N_TOTAL / 64), blkG, 0, stream>>>(x, W1, nullptr, dinv, big1, H1, IN_DIM, 0, 0);
  k_fill<<<dim3((unsigned)((size_t)N_TOTAL * H1 / 256)), blk, 0, stream>>>(big2, 0.0f, (size_t)N_TOTAL * H1);
  k_edge_agg<<<dim3(N_EDGES * 32 / 256), blk, 0, stream>>>(big1, src, dst, big2, N_EDGES, H1);
  k_gcn_final<<<dim3((unsigned)((size_t)N_TOTAL * H1 / 256)), blk, 0, stream>>>(big2, big1, dinv, b1, big1,
                                                                                 (size_t)N_TOTAL * H1, 8);
  // big1 now holds x1

  // --- attn block 1 --------------------------------------------------------
  k_gemm_wmma<<<dim3(H1 / 64, M_HOSTS / 64), blkG, 0, stream>>>(big1, a1W, a1b, nullptr, regA, H1, H1, 1, 0);
  k_gemm_wmma<<<dim3(H1 / 64, M_HOSTS / 64), blkG, 0, stream>>>(big1, f1W, f1b, nullptr, regF, H1, H1, 1, 0);
  k_fill<<<dim3(64), blk, 0, stream>>>(aout, 0.0f, (size_t)BGRAPH * H1);
  k_attn_reduce<<<dim3(M_HOSTS * 32 / 256), blk, 0, stream>>>(regA, regF, aout, M_HOSTS, H1);
  k_g_update<<<dim3(BGRAPH * GDIM / 256), blk, 0, stream>>>(aout, g0, g1W, g1b, g1, H1);

  // --- GCN layer 2 ---------------------------------------------------------
  k_gemm_wmma<<<dim3(H2 / 64, N_TOTAL / 64), blkG, 0, stream>>>(big1, W2, nullptr, dinv, hs2, H2, H1, 0, 0);
  k_fill<<<dim3((unsigned)((size_t)N_TOTAL * H2 / 256)), blk, 0, stream>>>(acc2, 0.0f, (size_t)N_TOTAL * H2);
  k_edge_agg<<<dim3(N_EDGES * 32 / 256), blk, 0, stream>>>(hs2, src, dst, acc2, N_EDGES, H2);
  k_gcn_final<<<dim3((unsigned)((size_t)N_TOTAL * H2 / 256)), blk, 0, stream>>>(acc2, hs2, dinv, b2, hs2,
                                                                                 (size_t)N_TOTAL * H2, 6);
  // hs2 now holds x2

  // --- attn block 2 --------------------------------------------------------
  k_gemm_wmma<<<dim3(H2 / 64, M_HOSTS / 64), blkG, 0, stream>>>(hs2, a2W, a2b, nullptr, regA, H2, H2, 1, 0);
  k_gemm_wmma<<<dim3(H2 / 64, M_HOSTS / 64), blkG, 0, stream>>>(hs2, f2W, f2b, nullptr, regF, H2, H2, 1, 0);
  k_fill<<<dim3(16), blk, 0, stream>>>(aout, 0.0f, (size_t)BGRAPH * H2);
  k_attn_reduce<<<dim3(M_HOSTS * 32 / 256), blk, 0, stream>>>(regA, regF, aout, M_HOSTS, H2);
  k_g_update<<<dim3(BGRAPH * GDIM / 256), blk, 0, stream>>>(aout, g1, g2W, g2b, g2, H2);

  // --- output MLP + per-graph softmax --------------------------------------
  k_build_hostz<<<dim3((unsigned)(((size_t)M_HOSTS * (H2 + GDIM)) / 256)), blk, 0, stream>>>(hs2, g2, big1);
  k_gemm_wmma<<<dim3(H1 / 64, M_HOSTS / 64), blkG, 0, stream>>>(big1, Wo1, bo1, nullptr, regA, H1, H2 + GDIM, 0, 1);
  k_actions<<<dim3((M_HOSTS * N_ACT + 255) / 256), blk, 0, stream>>>(regA, Wo2, bo2, regF);
  k_out_softmax<<<dim3(BGRAPH), blk, 0, stream>>>(regF, out);
}